// HaarTransform_19748259627739
// MI455X (gfx1250) — compile-verified
//
#include <hip/hip_runtime.h>

// HaarTransform on (4,64,512,512) f32 -> 4x (4,64,256,256) f32.
// Bandwidth-bound (537 MB @ 23.3 TB/s ~= 23 us floor). The 2x2 butterfly is
// D = A_coef(16x4) x B_data(4x16) via V_WMMA_F32_16X16X4_F32: the natural
// coalesced float2 loads land exactly in the B-operand lane layout (lanes
// 0-15 = even row (a,b) pairs, lanes 16-31 = odd row (c,d) pairs), so no
// cross-lane shuffles are needed. NT hints keep the 537 MB stream out of L2.
// Round 2: batch 4 strips per step (4 clause-able loads -> 4 WMMAs -> one
// masked store region) for intra-wave MLP, and hoist all four output base
// pointers so stores use immediate offsets off fixed v-pairs.

typedef __attribute__((ext_vector_type(2))) float v2f;
typedef __attribute__((ext_vector_type(8))) float v8f;

namespace {
constexpr int kB = 4, kC = 64, kH = 512, kW = 512;
constexpr int kPlanes = kB * kC;            // 256 (B*C image planes)
constexpr int kH2 = kH / 2, kW2 = kW / 2;   // 256 x 256 outputs per plane
constexpr int kOutPlane = kH2 * kW2;        // 65536 elements per plane
constexpr size_t kOutStride = (size_t)kPlanes * kOutPlane; // 16,777,216 per output tensor
constexpr int kWavesPerBlock = 8;           // 256 threads = 8 wave32
constexpr int kRowPairs = kPlanes * kH2;    // 65536 total row-pairs
}

__global__ __launch_bounds__(256) void haar_wmma_kernel(const float* __restrict__ x,
                                                        float* __restrict__ out) {
  const int lane = threadIdx.x & 31;
  const int wave = threadIdx.x >> 5;
  const int rp   = blockIdx.x * kWavesPerBlock + wave;  // row-pair id
  const int plane = rp >> 8;                            // rp / kH2
  const int r     = rp & (kH2 - 1);                     // output row within plane

  // A operand: constant Haar coefficients, rows replicated mod 4.
  // A 16x4 layout: lane L<16 holds A[L][0..1]; lane L+16 holds A[L][2..3].
  // Row patterns (x0.5): Yl(+ + + +), xH(+ + - -), xV(+ - + -), xD(+ - - +).
  const int m = lane & 3;
  const bool hi = lane >= 16;
  v2f coef;
  if (!hi) {
    coef.x = 0.5f;                                   // coeff of a
    coef.y = (m & 2) ? -0.5f : 0.5f;                 // coeff of b
  } else {
    coef.x = (m & 1) ? -0.5f : 0.5f;                 // coeff of c
    coef.y = (((m >> 1) ^ m) & 1) ? -0.5f : 0.5f;    // coeff of d
  }

  // B operand source: lane n<16 reads (a,b) from even row at col 2n,
  // lane n+16 reads (c,d) from odd row at col 2n; each half is one 128B seg.
  const float* rowBase = x + (size_t)plane * kH * kW
                           + (size_t)(2 * r + (hi ? 1 : 0)) * kW
                           + (size_t)(lane & 15) * 2;

  // Four output base pointers, hoisted: stores become immediate offsets.
  float* pj[4];
  pj[0] = out + (size_t)plane * kOutPlane + (size_t)r * kW2 + lane;
  pj[1] = pj[0] + kOutStride;
  pj[2] = pj[0] + 2 * kOutStride;
  pj[3] = pj[0] + 3 * kOutStride;

  // 16 strips of 16 pixel-groups cover one output row; process 4 per batch.
  #pragma unroll
  for (int batch = 0; batch < 4; ++batch) {
    const float* lb = rowBase + batch * 128;   // 4 strips * 32 input floats

    v2f b[4];
    #pragma unroll
    for (int s = 0; s < 4; ++s)
      b[s] = __builtin_nontemporal_load((const v2f*)(lb + 32 * s));

    v8f a[4];
    #pragma unroll
    for (int s = 0; s < 4; ++s) {
      v8f acc = {};
      // 8 args: (neg_a, A, neg_b, B, c_mod, C, reuse_a, reuse_b)
      a[s] = __builtin_amdgcn_wmma_f32_16x16x4_f32(
          false, coef, false, b[s], (short)0, acc, false, false);
    }

    // D layout: VGPR j, lanes 0-15 = output type j for groups 0-15.
    if (lane < 16) {
      #pragma unroll
      for (int j = 0; j < 4; ++j) {
        float* q = pj[j] + batch * 64;         // 4 strips * 16 output floats
        #pragma unroll
        for (int s = 0; s < 4; ++s)
          __builtin_nontemporal_store(a[s][j], q + 16 * s);
      }
    }
  }
}

extern "C" void kernel_launch(void* const* d_in, const int* in_sizes, int n_in,
                              void* d_out, int out_size, void* d_ws, size_t ws_size,
                              hipStream_t stream) {
  (void)in_sizes; (void)n_in; (void)out_size; (void)d_ws; (void)ws_size;
  const float* x = (const float*)d_in[0];
  float* out = (float*)d_out;
  dim3 grid(kRowPairs / kWavesPerBlock);  // 8192 blocks
  dim3 block(kWavesPerBlock * 32);        // 256 threads = 8 wave32
  hipLaunchKernelGGL(haar_wmma_kernel, grid, block, 0, stream, x, out);
}